// MoEBlock_3959959847166
// MI455X (gfx1250) — compile-verified
//
#include <hip/hip_runtime.h>
#include <hip/hip_bf16.h>
#include <math.h>
#include <stdint.h>

typedef float v2f __attribute__((ext_vector_type(2)));
typedef float v8f __attribute__((ext_vector_type(8)));

#define D_MODEL 1024
#define N_EXP   8
#define D_FF    4096
#define TILE_M  32
#define FC      128      // F-chunk columns per pipeline stage
#define HPAD    132      // FC + 4 padding -> conflict-free ds_load_b64
#define MAXTOK  4096
#define KBLK    32       // K rows per async-staged w1 panel

// D = A(16x4 f32) * B(4x16 f32) + C(16x16 f32), wave32
#define WMMA_F32(A, B, C) \
  __builtin_amdgcn_wmma_f32_16x16x4_f32(false, (A), false, (B), (short)0, (C), false, false)

#define WAIT_ASYNC4 asm volatile("s_wait_asynccnt 0x4" ::: "memory")
#define WAIT_ASYNC0 asm volatile("s_wait_asynccnt 0x0" ::: "memory")

__device__ __forceinline__ float gelu_exact(float v) {
  return 0.5f * v * (1.0f + erff(v * 0.70710678118654752440f));
}

// Async-DMA one 32x16 fp32 panel of w1 (one wave's columns for one K-block)
// into this wave's private LDS double-buffer. 4 x b128 per lane-set:
// lane -> (row = lane>>2, 4 cols = (lane&3)*4), kk steps 8 rows each.
__device__ __forceinline__ void issue_w1_block(const float* __restrict__ gbase,
                                               uint32_t lds_wave_base,
                                               int row0, int colbase, int lane, int buf) {
  const int r  = lane >> 2;          // 0..7
  const int c4 = (lane & 3) << 2;    // 0,4,8,12
  uint32_t dst = lds_wave_base + (uint32_t)(buf * 2048 + r * 64 + ((lane & 3) << 4));
  uint32_t src = (uint32_t)((((row0 + r) * D_FF) + colbase + c4) << 2);
#pragma unroll
  for (int kk = 0; kk < 4; ++kk) {
    asm volatile("global_load_async_to_lds_b128 %0, %1, %2"
                 :: "v"(dst + (uint32_t)(kk * 512)),
                    "v"(src + (uint32_t)(kk * 8 * D_FF * 4)),
                    "s"(gbase)
                 : "memory");
  }
}

// ---------------------------------------------------------------------------
// Zero the accumulation output region, the expert_counts output, and the
// per-expert workspace counters (must run every call; we accumulate atomics).
// ---------------------------------------------------------------------------
__global__ void init_kernel(float* __restrict__ out, float* __restrict__ counts_f,
                            int* __restrict__ ws_cnt, int n_out) {
  int i = blockIdx.x * blockDim.x + threadIdx.x;
  if (i < n_out) out[i] = 0.0f;
  if (i < N_EXP) { counts_f[i] = 0.0f; ws_cnt[i] = 0; }
}

// ---------------------------------------------------------------------------
// Router: logits -> softmax -> top-2 -> dense dispatch mask + expert counts
// + per-expert gather lists (token index, combine weight) in workspace.
// ---------------------------------------------------------------------------
__global__ void router_kernel(const float* __restrict__ x,
                              const float* __restrict__ gw,   // [D, E]
                              const float* __restrict__ gb,   // [E]
                              float* __restrict__ dispatch,   // [n_tok, E]
                              float* __restrict__ counts_f,   // [E]
                              int*   __restrict__ ws_cnt,     // [E]
                              int*   __restrict__ ws_tok,     // [E, MAXTOK]
                              float* __restrict__ ws_wgt,     // [E, MAXTOK]
                              int n_tok) {
  int t = blockIdx.x * blockDim.x + threadIdx.x;
  if (t >= n_tok) return;

  float logit[N_EXP];
#pragma unroll
  for (int e = 0; e < N_EXP; ++e) logit[e] = gb[e];

  const float* xr = x + (size_t)t * D_MODEL;
  for (int d = 0; d < D_MODEL; ++d) {
    float xv = xr[d];
    const float* gr = gw + (size_t)d * N_EXP;
#pragma unroll
    for (int e = 0; e < N_EXP; ++e) logit[e] = fmaf(xv, gr[e], logit[e]);
  }

  float mx = logit[0];
#pragma unroll
  for (int e = 1; e < N_EXP; ++e) mx = fmaxf(mx, logit[e]);
  float sum = 0.0f;
  float sc[N_EXP];
#pragma unroll
  for (int e = 0; e < N_EXP; ++e) { sc[e] = __expf(logit[e] - mx); sum += sc[e]; }
  float inv = 1.0f / sum;
#pragma unroll
  for (int e = 0; e < N_EXP; ++e) sc[e] *= inv;

  // top-2 (ties -> lowest index, matching jax.lax.top_k)
  int i1 = 0;
#pragma unroll
  for (int e = 1; e < N_EXP; ++e) if (sc[e] > sc[i1]) i1 = e;
  int i2 = (i1 == 0) ? 1 : 0;
#pragma unroll
  for (int e = 0; e < N_EXP; ++e)
    if (e != i1 && sc[e] > sc[i2]) i2 = e;

  float* dr = dispatch + (size_t)t * N_EXP;
#pragma unroll
  for (int e = 0; e < N_EXP; ++e)
    dr[e] = (e == i1 || e == i2) ? sc[e] : 0.0f;

  atomicAdd(&counts_f[i1], sc[i1]);
  atomicAdd(&counts_f[i2], sc[i2]);

  int p1 = atomicAdd(&ws_cnt[i1], 1);
  ws_tok[i1 * MAXTOK + p1] = t;
  ws_wgt[i1 * MAXTOK + p1] = sc[i1];
  int p2 = atomicAdd(&ws_cnt[i2], 1);
  ws_tok[i2 * MAXTOK + p2] = t;
  ws_wgt[i2 * MAXTOK + p2] = sc[i2];
}

// ---------------------------------------------------------------------------
// Fused expert FFN, tile = 32 gathered tokens, 8 wave32s.
// GEMM1: acc += x_tile @ w1-panel with w1 panels async-DMA'd into per-wave
//        LDS double buffers (global_load_async_to_lds_b128 + s_wait_asynccnt).
// h-chunk (gelu) staged in padded LDS; GEMM2 accumulates [32x1024] in
// 128 VGPRs of f32 WMMA accumulators; weighted atomic scatter into out.
// ---------------------------------------------------------------------------
__global__ __launch_bounds__(256) void expert_ffn_kernel(
    const float* __restrict__ x,
    const float* __restrict__ w1, const float* __restrict__ b1,
    const float* __restrict__ w2, const float* __restrict__ b2,
    const int* __restrict__ ws_cnt, const int* __restrict__ ws_tok,
    const float* __restrict__ ws_wgt,
    float* __restrict__ out) {
  __shared__ __align__(16) float lds_w1[8 * 2 * KBLK * 16]; // 32 KB: 8 waves x 2 bufs x 32x16
  __shared__ __align__(16) float hsm[TILE_M * HPAD];        // ~16.9 KB h-chunk
  __shared__ int   s_tok[TILE_M];
  __shared__ float s_w[TILE_M];

  const int e     = blockIdx.y;
  const int count = ws_cnt[e];
  const int m0    = blockIdx.x * TILE_M;
  if (m0 >= count) return;               // block-uniform exit

  const int tid = threadIdx.x;
  if (tid < TILE_M) {
    int idx = m0 + tid;
    int ok  = idx < count;
    int src = ok ? idx : (count - 1);    // duplicate last valid row, weight 0
    s_tok[tid] = ws_tok[e * MAXTOK + src];
    s_w[tid]   = ok ? ws_wgt[e * MAXTOK + src] : 0.0f;
  }
  __syncthreads();

  const int lane = tid & 31;
  const int wid  = tid >> 5;
  const int l16  = lane & 15;
  const int half = lane >> 4;
  const int koff = 2 * half;             // K offset for this lane half

  const float* w1e = w1 + (size_t)e * D_MODEL * D_FF;
  const float* w2e = w2 + (size_t)e * D_FF * D_MODEL;
  const float* b1e = b1 + (size_t)e * D_FF;
  const float* b2e = b2 + (size_t)e * D_MODEL;

  // Per-lane x row pointers for the two M-subtiles (A fragments).
  const float* px0 = x + (size_t)s_tok[l16]      * D_MODEL;
  const float* px1 = x + (size_t)s_tok[16 + l16] * D_MODEL;

  // This wave's private staging region (LDS byte offset of low 32 bits).
  const uint32_t w1_lds_base = (uint32_t)(uintptr_t)&lds_w1[wid * (2 * KBLK * 16)];
  const float*   w1_lds      = &lds_w1[wid * (2 * KBLK * 16)];

  // GEMM2 accumulators: [2 M-subtiles][8 N-subtiles] = 128 VGPRs/lane
  v8f acc2[2][8];
#pragma unroll
  for (int i = 0; i < 2; ++i)
#pragma unroll
    for (int j = 0; j < 8; ++j) acc2[i][j] = (v8f)0.0f;

  for (int fcb = 0; fcb < D_FF; fcb += FC) {
    const int colbase = fcb + wid * 16;     // this wave's 16 w1 columns

    // ---- GEMM1: h_chunk[32 x FC] = x_tile @ w1[:, chunk], async-staged B
    v8f acc1[2];
    acc1[0] = (v8f)0.0f;
    acc1[1] = (v8f)0.0f;

    issue_w1_block(w1e, w1_lds_base, 0,    colbase, lane, 0);
    issue_w1_block(w1e, w1_lds_base, KBLK, colbase, lane, 1);

#pragma unroll 1
    for (int kb = 0; kb < D_MODEL; kb += KBLK) {
      const int buf = (kb >> 5) & 1;
      if (kb + KBLK >= D_MODEL) { WAIT_ASYNC0; } else { WAIT_ASYNC4; }
      const float* wb = w1_lds + buf * (KBLK * 16);
#pragma unroll
      for (int ks = 0; ks < KBLK; ks += 4) {
        v2f a0 = *(const v2f*)(px0 + kb + ks + koff);
        v2f a1 = *(const v2f*)(px1 + kb + ks + koff);
        v2f b;
        b.x = wb[(ks + koff) * 16 + l16];
        b.y = wb[(ks + koff + 1) * 16 + l16];
        acc1[0] = WMMA_F32(a0, b, acc1[0]);
        acc1[1] = WMMA_F32(a1, b, acc1[1]);
      }
      if (kb + 2 * KBLK < D_MODEL)
        issue_w1_block(w1e, w1_lds_base, kb + 2 * KBLK, colbase, lane, buf);
    }

    __syncthreads();   // previous chunk's h reads complete before overwrite
    {
      int lcol = wid * 16 + l16;
      float bias = b1e[fcb + lcol];
#pragma unroll
      for (int ms = 0; ms < 2; ++ms)
#pragma unroll
        for (int r = 0; r < 8; ++r) {
          int ml = ms * 16 + half * 8 + r;   // C/D layout: M = r + 8*half
          hsm[ml * HPAD + lcol] = gelu_exact(acc1[ms][r] + bias);
        }
    }
    __syncthreads();

    // ---- GEMM2: y[32 x 1024] += h_chunk[32 x FC] @ w2[chunk, 1024]
    const int n2base = wid * 128 + l16;       // wave covers 128 of 1024 cols
    for (int kb2 = 0; kb2 < FC; kb2 += 4) {
      v2f a0 = *(const v2f*)&hsm[l16 * HPAD + kb2 + koff];
      v2f a1 = *(const v2f*)&hsm[(16 + l16) * HPAD + kb2 + koff];
      const float* bp2 = w2e + (size_t)(fcb + kb2 + koff) * D_MODEL + n2base;
#pragma unroll
      for (int nt2 = 0; nt2 < 8; ++nt2) {
        v2f b;
        b.x = bp2[nt2 * 16];
        b.y = bp2[nt2 * 16 + D_MODEL];
        acc2[0][nt2] = WMMA_F32(a0, b, acc2[0][nt2]);
        acc2[1][nt2] = WMMA_F32(a1, b, acc2[1][nt2]);
      }
    }
  }

  // ---- Epilogue: out[tok, :] += weight * (y + b2), scatter with f32 atomics
#pragma unroll
  for (int ms = 0; ms < 2; ++ms)
#pragma unroll
    for (int nt2 = 0; nt2 < 8; ++nt2) {
      int col = wid * 128 + nt2 * 16 + l16;
      float bias = b2e[col];
#pragma unroll
      for (int r = 0; r < 8; ++r) {
        int ml = ms * 16 + half * 8 + r;
        float w = s_w[ml];
        if (w != 0.0f) {
          int tok = s_tok[ml];
          atomicAdd(&out[(size_t)tok * D_MODEL + col],
                    w * (acc2[ms][nt2][r] + bias));
        }
      }
    }
}

// ---------------------------------------------------------------------------
extern "C" void kernel_launch(void* const* d_in, const int* in_sizes, int n_in,
                              void* d_out, int out_size, void* d_ws, size_t ws_size,
                              hipStream_t stream) {
  const float* x  = (const float*)d_in[0];
  const float* gw = (const float*)d_in[1];
  const float* gb = (const float*)d_in[2];
  const float* w1 = (const float*)d_in[3];
  const float* b1 = (const float*)d_in[4];
  const float* w2 = (const float*)d_in[5];
  const float* b2 = (const float*)d_in[6];

  const int n_tok = in_sizes[0] / D_MODEL;   // B*S = 4096

  float* out_f    = (float*)d_out;                         // [n_tok, D]
  float* dispatch = out_f + (size_t)n_tok * D_MODEL;       // [n_tok, E]
  float* counts_f = dispatch + (size_t)n_tok * N_EXP;      // [E]

  int*   ws_cnt = (int*)d_ws;                              // [E] (padded)
  int*   ws_tok = ws_cnt + 16;                             // [E, MAXTOK]
  float* ws_wgt = (float*)(ws_tok + N_EXP * MAXTOK);       // [E, MAXTOK]

  const int n_out = n_tok * D_MODEL;
  init_kernel<<<(n_out + 255) / 256, 256, 0, stream>>>(out_f, counts_f, ws_cnt, n_out);

  router_kernel<<<(n_tok + 255) / 256, 256, 0, stream>>>(
      x, gw, gb, dispatch, counts_f, ws_cnt, ws_tok, ws_wgt, n_tok);

  dim3 grid((n_tok + TILE_M - 1) / TILE_M, N_EXP);
  expert_ffn_kernel<<<grid, 256, 0, stream>>>(
      x, w1, b1, w2, b2, ws_cnt, ws_tok, ws_wgt, out_f);
}